// CausalEdgeAttention_58025008169388
// MI455X (gfx1250) — compile-verified
//
#include <hip/hip_runtime.h>

typedef __attribute__((ext_vector_type(2))) float v2f;
typedef __attribute__((ext_vector_type(8))) float v8f;

__device__ __forceinline__ v8f wmma4(v2f a, v2f b, v8f c) {
  // V_WMMA_F32_16X16X4_F32: exact f32 16x16 tile MACs, wave32
  return __builtin_amdgcn_wmma_f32_16x16x4_f32(false, a, false, b, (short)0, c,
                                               false, false);
}

// ---------------------------------------------------------------------------
// Generic 16x16-tile f32 WMMA GEMM: C[M x Nreal] = A[M x K] @ B[K x Nreal]
// One wave per block/tile. M multiple of 16, K multiple of 4, Nreal <= 16*ntilesX.
// Branch-free inner loop: out-of-range columns use a clamped address and a
// 0.0 multiplier (v_cndmask select, no EXEC divergence around the WMMAs).
// ---------------------------------------------------------------------------
__global__ void gemm_tile_wmma(const float* __restrict__ A,
                               const float* __restrict__ B,
                               float* __restrict__ C,
                               int K, int Nreal, int ntilesX) {
  const int lane = threadIdx.x & 31;
  const int mn   = lane & 15;   // row (A) / col (B,C) within tile
  const int hf   = lane >> 4;   // half-wave select
  const int kb   = hf * 2;      // K sub-offset for A/B operands
  const int tile = blockIdx.x;
  const int tn = tile % ntilesX;
  const int tm = tile / ntilesX;
  const int m  = tm * 16 + mn;
  const int n  = tn * 16 + mn;
  const int nc = min(n, Nreal - 1);               // clamped (always valid) col
  const float msk = (n < Nreal) ? 1.0f : 0.0f;    // zero pad columns
  v8f acc = {};
  for (int k0 = 0; k0 < K; k0 += 4) {
    v2f a, b;
    a.x = A[m * K + k0 + kb];
    a.y = A[m * K + k0 + kb + 1];
    b.x = B[(k0 + kb) * Nreal + nc] * msk;
    b.y = B[(k0 + kb + 1) * Nreal + nc] * msk;
    acc = wmma4(a, b, acc);
  }
  if (n < Nreal) {
#pragma unroll
    for (int v = 0; v < 8; ++v) {
      int row = tm * 16 + v + 8 * hf;  // C-layout: VGPR v holds rows v, v+8
      C[row * Nreal + n] = acc[v];
    }
  }
}

__global__ void zero512(float* __restrict__ p) {
  if (threadIdx.x < 512) p[threadIdx.x] = 0.0f;
}

// ---------------------------------------------------------------------------
// Stage A: per-channel sum / sum-of-squares of h1 = relu(x @ W1 + b1) over all
// E edges, x = 0.5*(nf[src]+nf[tgt]). Each wave processes 16 edges per tile
// via two 16x16x4 WMMA steps (K=8) per 16-channel column tile. Per-channel
// partials are kept in registers across the grid-stride loop (lane l owns
// channels ct*16 + (l&15)); LDS/global atomics only at flush time.
// ---------------------------------------------------------------------------
__global__ void stage_stats(const float* __restrict__ nf,
                            const int* __restrict__ eidx,
                            const float* __restrict__ W1,
                            const float* __restrict__ nb1,
                            float* __restrict__ gsum, float* __restrict__ gsq,
                            int E, int numTiles) {
  __shared__ float sSum[256];
  __shared__ float sSq[256];
  for (int i = threadIdx.x; i < 256; i += blockDim.x) { sSum[i] = 0.f; sSq[i] = 0.f; }
  __syncthreads();

  const int lane = threadIdx.x & 31;
  const int mn   = lane & 15;
  const int hf   = lane >> 4;
  const int kb   = hf * 2;
  const int waveId   = (blockIdx.x * blockDim.x + threadIdx.x) >> 5;
  const int numWaves = (gridDim.x * blockDim.x) >> 5;

  float accS[16], accQ[16];
#pragma unroll
  for (int ct = 0; ct < 16; ++ct) { accS[ct] = 0.f; accQ[ct] = 0.f; }

  for (int t = waveId; t < numTiles; t += numWaves) {
    const int base = t * 16;
    const int e = min(base + mn, E - 1);
    const int s = eidx[e];
    const int d = eidx[E + e];
    const float* ps = nf + s * 8;
    const float* pt = nf + d * 8;
    v2f a0, a1;  // A operand: a[j] = x[m][kb+j] (kt=0) and x[m][4+kb+j] (kt=1)
    a0.x = 0.5f * (ps[kb]     + pt[kb]);
    a0.y = 0.5f * (ps[kb + 1] + pt[kb + 1]);
    a1.x = 0.5f * (ps[4 + kb]     + pt[4 + kb]);
    a1.y = 0.5f * (ps[4 + kb + 1] + pt[4 + kb + 1]);

#pragma unroll
    for (int ct = 0; ct < 16; ++ct) {
      const int col = ct * 16 + mn;
      v2f b0, b1;  // loop-invariant across t: hoisted to registers by LICM
      b0.x = W1[(kb)     * 256 + col];
      b0.y = W1[(kb + 1) * 256 + col];
      b1.x = W1[(4 + kb) * 256 + col];
      b1.y = W1[(5 + kb) * 256 + col];
      v8f c = {};
      c = wmma4(a0, b0, c);
      c = wmma4(a1, b1, c);
      const float bias = nb1[col];
#pragma unroll
      for (int v = 0; v < 8; ++v) {
        const int row = base + v + 8 * hf;
        float h = fmaxf(c[v] + bias, 0.0f);
        h = (row < E) ? h : 0.0f;  // mask tail rows
        accS[ct] += h;
        accQ[ct] += h * h;
      }
    }
  }

  // flush: both half-waves hit the same channel -> LDS atomics, then global
#pragma unroll
  for (int ct = 0; ct < 16; ++ct) {
    atomicAdd(&sSum[ct * 16 + mn], accS[ct]);
    atomicAdd(&sSq[ct * 16 + mn], accQ[ct]);
  }
  __syncthreads();
  for (int i = threadIdx.x; i < 256; i += blockDim.x) {
    atomicAdd(&gsum[i], sSum[i]);
    atomicAdd(&gsq[i], sSq[i]);
  }
}

// ---------------------------------------------------------------------------
// Stage B (1 block, 256 threads): fold BatchNorm + linear chain.
//   a = gamma*rsqrt(var+eps), c = beta - mu*a
//   Wfold[k][d] = 0.5 * a[k] * (W2@Wv@Wo@Wp)[k][d]   (M8 precomputed)
//   bfold[d]    = 0.5 * ((((c@W2+b2)@Wv+bv)@Wo+bo)@Wp+bp)[d]
// ---------------------------------------------------------------------------
__global__ void stage_fold(const float* __restrict__ gsum, const float* __restrict__ gsq,
                           const float* __restrict__ M8,
                           const float* __restrict__ gamma, const float* __restrict__ beta,
                           const float* __restrict__ W2, const float* __restrict__ b2,
                           const float* __restrict__ Wv, const float* __restrict__ bv,
                           const float* __restrict__ Wo, const float* __restrict__ bo,
                           const float* __restrict__ Wp, const float* __restrict__ bp,
                           float* __restrict__ Wfold, float* __restrict__ bfold, int E) {
  __shared__ float sc[256], su[256], sv[256], sw[256];
  const int k = threadIdx.x;
  const float inv = 1.0f / (float)E;
  const float mu  = gsum[k] * inv;
  const float var = gsq[k] * inv - mu * mu;
  const float aK  = gamma[k] * rsqrtf(var + 1e-5f);
  const float cK  = beta[k] - mu * aK;
  sc[k] = cK;
#pragma unroll
  for (int d = 0; d < 8; ++d) Wfold[k * 8 + d] = 0.5f * aK * M8[k * 8 + d];
  __syncthreads();
  float u = b2[k];
  for (int i = 0; i < 256; ++i) u += sc[i] * W2[i * 256 + k];
  su[k] = u;
  __syncthreads();
  float vv = bv[k];
  for (int i = 0; i < 256; ++i) vv += su[i] * Wv[i * 256 + k];
  sv[k] = vv;
  __syncthreads();
  float w = bo[k];
  for (int i = 0; i < 256; ++i) w += sv[i] * Wo[i * 256 + k];
  sw[k] = w;
  __syncthreads();
  if (k < 8) {
    float bb = bp[k];
    for (int i = 0; i < 256; ++i) bb += sw[i] * Wp[i * 8 + k];
    bfold[k] = 0.5f * bb;
  }
}

// ---------------------------------------------------------------------------
// Stage C: out = edge_attr + relu(x@W1+b1) @ Wfold + bfold.
// Per wave: 16 edges. GEMM1 (K=8) via WMMA into C-layout, stage h1 through
// per-wave LDS (transpose to A-layout), then GEMM2 (K=256, 64 WMMA steps)
// against Wfold^T held in LDS (cols 8..15 zero-padded).
// ---------------------------------------------------------------------------
#define H1S 258  // padded row stride (floats) -> conflict-free half-wave banks
__global__ void stage_apply(const float* __restrict__ edge_attr,
                            const float* __restrict__ nf,
                            const int* __restrict__ eidx,
                            const float* __restrict__ W1,
                            const float* __restrict__ nb1,
                            const float* __restrict__ Wfold,
                            const float* __restrict__ bfold,
                            float* __restrict__ out, int E, int numTiles) {
  __shared__ float sWT[16 * H1S];      // Wfold transposed: sWT[n*H1S + k]
  __shared__ float sH1[2][16 * H1S];   // per-wave h1 staging (2 waves/block)
  for (int i = threadIdx.x; i < 16 * 256; i += blockDim.x) {
    const int col = i >> 8;     // output dim 0..15
    const int kk  = i & 255;
    sWT[col * H1S + kk] = (col < 8) ? Wfold[kk * 8 + col] : 0.0f;
  }
  __syncthreads();

  const int lane = threadIdx.x & 31;
  const int mn   = lane & 15;
  const int hf   = lane >> 4;
  const int kb   = hf * 2;
  float* myH1 = sH1[threadIdx.x >> 5];
  const int waveId   = (blockIdx.x * blockDim.x + threadIdx.x) >> 5;
  const int numWaves = (gridDim.x * blockDim.x) >> 5;

  for (int t = waveId; t < numTiles; t += numWaves) {
    const int base = t * 16;
    const int e = min(base + mn, E - 1);
    const int s = eidx[e];
    const int d = eidx[E + e];
    const float* ps = nf + s * 8;
    const float* pt = nf + d * 8;
    v2f a0, a1;
    a0.x = 0.5f * (ps[kb]     + pt[kb]);
    a0.y = 0.5f * (ps[kb + 1] + pt[kb + 1]);
    a1.x = 0.5f * (ps[4 + kb]     + pt[4 + kb]);
    a1.y = 0.5f * (ps[4 + kb + 1] + pt[4 + kb + 1]);

    // GEMM1: h1 tile [16 x 256] -> LDS (transposed to row-major per edge)
    for (int ct = 0; ct < 16; ++ct) {
      const int col = ct * 16 + mn;
      v2f b0, b1;
      b0.x = W1[(kb)     * 256 + col];
      b0.y = W1[(kb + 1) * 256 + col];
      b1.x = W1[(4 + kb) * 256 + col];
      b1.y = W1[(5 + kb) * 256 + col];
      v8f c = {};
      c = wmma4(a0, b0, c);
      c = wmma4(a1, b1, c);
      const float bias = nb1[col];
#pragma unroll
      for (int v = 0; v < 8; ++v) {
        myH1[(v + 8 * hf) * H1S + col] = fmaxf(c[v] + bias, 0.0f);
      }
    }
    // Intra-wave cross-lane LDS hazard: wait for our ds_stores before reading
    __builtin_amdgcn_wave_barrier();
    asm volatile("s_wait_dscnt 0" ::: "memory");
    __builtin_amdgcn_wave_barrier();

    // GEMM2: [16 x 256] @ [256 x 16(pad)] -> 16x16, K in 64 steps of 4
    v8f acc = {};
    for (int kt = 0; kt < 64; ++kt) {
      const int k0 = kt * 4 + kb;
      v2f a, b;
      a.x = myH1[mn * H1S + k0];       // A: h1[m][k]
      a.y = myH1[mn * H1S + k0 + 1];
      b.x = sWT[mn * H1S + k0];        // B: Wfold[k][n] (transposed store)
      b.y = sWT[mn * H1S + k0 + 1];
      acc = wmma4(a, b, acc);
    }

    if (mn < 8) {
      const float bf = bfold[mn];
#pragma unroll
      for (int v = 0; v < 8; ++v) {
        const int row = base + v + 8 * hf;
        if (row < E) {
          const int o = row * 8 + mn;
          out[o] = edge_attr[o] + acc[v] + bf;
        }
      }
    }
  }
}

extern "C" void kernel_launch(void* const* d_in, const int* in_sizes, int n_in,
                              void* d_out, int out_size, void* d_ws, size_t ws_size,
                              hipStream_t stream) {
  (void)n_in; (void)out_size; (void)ws_size;
  const float* edge_attr = (const float*)d_in[0];
  const float* nf        = (const float*)d_in[1];
  const int*   eidx      = (const int*)d_in[2];
  // d_in[3..8] = edge-encoder params: provably dead in the reference output
  const float* nW1   = (const float*)d_in[9];
  const float* nb1   = (const float*)d_in[10];
  const float* ngam  = (const float*)d_in[11];
  const float* nbet  = (const float*)d_in[12];
  const float* nW2   = (const float*)d_in[13];
  const float* nb2   = (const float*)d_in[14];
  const float* Wv    = (const float*)d_in[15];
  const float* bv    = (const float*)d_in[16];
  const float* Wo    = (const float*)d_in[17];
  const float* bo    = (const float*)d_in[18];
  const float* Wp    = (const float*)d_in[19];
  const float* bp    = (const float*)d_in[20];

  const int E = in_sizes[2] / 2;
  const int numTiles = (E + 15) / 16;

  float* ws    = (float*)d_ws;
  float* T1    = ws;                 // 256*256
  float* T2    = T1 + 65536;         // 256*256
  float* M8    = T2 + 65536;         // 256*8
  float* gsum  = M8 + 2048;          // 256
  float* gsq   = gsum + 256;         // 256
  float* Wfold = gsq + 256;          // 256*8
  float* bfold = Wfold + 2048;       // 8

  // One-time weight chain: M8 = W2 @ Wv @ Wo @ Wp (all f32 WMMA)
  gemm_tile_wmma<<<256, 32, 0, stream>>>(nW2, Wv, T1, 256, 256, 16);
  gemm_tile_wmma<<<256, 32, 0, stream>>>(T1, Wo, T2, 256, 256, 16);
  gemm_tile_wmma<<<16, 32, 0, stream>>>(T2, Wp, M8, 256, 8, 1);

  zero512<<<1, 512, 0, stream>>>(gsum);  // zeros gsum[256] + gsq[256]

  stage_stats<<<512, 256, 0, stream>>>(nf, eidx, nW1, nb1, gsum, gsq, E, numTiles);
  stage_fold<<<1, 256, 0, stream>>>(gsum, gsq, M8, ngam, nbet, nW2, nb2,
                                    Wv, bv, Wo, bo, Wp, bp, Wfold, bfold, E);
  stage_apply<<<2048, 64, 0, stream>>>(edge_attr, nf, eidx, nW1, nb1,
                                       Wfold, bfold, (float*)d_out, E, numTiles);
}